// LinTransformer_8014408974649
// MI455X (gfx1250) — compile-verified
//
#include <hip/hip_runtime.h>

// MI455X / gfx1250, wave32. All matrix math via V_WMMA_F32_16X16X4_F32.
// B=8, N=2048, D=1024, DK=64. Linear attention reassociated so that neither
// the NxN score matrix, V, nor the attention tensor is ever materialized.

typedef __attribute__((ext_vector_type(2))) float v2f;
typedef __attribute__((ext_vector_type(8))) float v8f;

#define BB 8
#define NN 2048
#define DD 1024
#define KK 64

__device__ __forceinline__ v8f wmma4(v2f a, v2f b, v8f c) {
  // 8 args: (neg_a, A, neg_b, B, c_mod, C, reuse_a, reuse_b)
  return __builtin_amdgcn_wmma_f32_16x16x4_f32(false, a, false, b, (short)0, c,
                                               false, false);
}

// A-fragment of a row-major matrix, or B-fragment built from W^T where W is
// [out,in] row-major: element [l, 2h+j] lives at p[l*stride + 2h + j].
__device__ __forceinline__ v2f frag_contig(const float* p, int stride) {
  int l = threadIdx.x & 15;
  int h = (threadIdx.x >> 4) & 1;
  const float* q = p + (size_t)l * stride + h * 2;
  v2f r;
  r.x = q[0];
  r.y = q[1];
  return r;
}

// B-fragment of a row-major [K,N] matrix, or A-fragment of a transposed
// (column-major) operand: element [2h+j, l] lives at p[(2h+j)*stride + l].
__device__ __forceinline__ v2f frag_strided(const float* p, int stride) {
  int l = threadIdx.x & 15;
  int h = (threadIdx.x >> 4) & 1;
  v2f r;
  r.x = p[(size_t)(h * 2 + 0) * stride + l];
  r.y = p[(size_t)(h * 2 + 1) * stride + l];
  return r;
}

// C/D 16x16 tile: lane l holds column l%16; VGPR v holds row v + 8*(l/16).
__device__ __forceinline__ void store_tile(float* p, int stride, v8f c) {
  int l = threadIdx.x & 15;
  int h = (threadIdx.x >> 4) & 1;
  float* q = p + (size_t)(h * 8) * stride + l;
#pragma unroll
  for (int v = 0; v < 8; ++v) q[(size_t)v * stride] = c[v];
}

__device__ __forceinline__ v8f load_tile(const float* p, int stride) {
  int l = threadIdx.x & 15;
  int h = (threadIdx.x >> 4) & 1;
  const float* q = p + (size_t)(h * 8) * stride + l;
  v8f c;
#pragma unroll
  for (int v = 0; v < 8; ++v) c[v] = q[(size_t)v * stride];
  return c;
}

// K1: Q = x@Wq^T, K = x@Wk^T, A1 = x@W1^T   ([B,N,64] each)
// grid: B*(N/16) blocks of 128; wave w computes columns [16w,16w+16) of all 3.
__global__ void __launch_bounds__(128) qka_kernel(
    const float* __restrict__ x, const float* __restrict__ Wq,
    const float* __restrict__ Wk, const float* __restrict__ W1,
    float* __restrict__ Q, float* __restrict__ Km, float* __restrict__ A1) {
  int wave = threadIdx.x >> 5;
  int b = blockIdx.x / (NN / 16);
  int row0 = (blockIdx.x % (NN / 16)) * 16;
  int col0 = wave * 16;
  const float* xb = x + (size_t)b * NN * DD + (size_t)row0 * DD;
  const float* wq = Wq + (size_t)col0 * DD;
  const float* wk = Wk + (size_t)col0 * DD;
  const float* w1 = W1 + (size_t)col0 * DD;
  v8f cq = {}, ck = {}, c1 = {};
  for (int k0 = 0; k0 < DD; k0 += 4) {
    v2f a = frag_contig(xb + k0, DD);
    cq = wmma4(a, frag_contig(wq + k0, DD), cq);
    ck = wmma4(a, frag_contig(wk + k0, DD), ck);
    c1 = wmma4(a, frag_contig(w1 + k0, DD), c1);
  }
  size_t o = (size_t)b * NN * KK + (size_t)row0 * KK + col0;
  store_tile(Q + o, KK, cq);
  store_tile(Km + o, KK, ck);
  store_tile(A1 + o, KK, c1);
}

// K2: S[b] = K[b]^T @ x[b]   ([B,64,1024]); reduction over N=2048.
__global__ void __launch_bounds__(128) s_kernel(const float* __restrict__ x,
                                                const float* __restrict__ Km,
                                                float* __restrict__ S) {
  int wave = threadIdx.x >> 5;
  int b = blockIdx.x / (DD / 16);
  int d0 = (blockIdx.x % (DD / 16)) * 16;
  int m0 = wave * 16;
  const float* kb = Km + (size_t)b * NN * KK + m0;  // A = K^T (col-major view)
  const float* xb = x + (size_t)b * NN * DD + d0;   // B = x  (row-major)
  v8f c = {};
  for (int n0 = 0; n0 < NN; n0 += 4) {
    v2f a = frag_strided(kb + (size_t)n0 * KK, KK);
    v2f bb = frag_strided(xb + (size_t)n0 * DD, DD);
    c = wmma4(a, bb, c);
  }
  store_tile(S + (size_t)b * KK * DD + (size_t)m0 * DD + d0, DD, c);
}

// K3a: KV[b] = S[b] @ Wv^T   ([B,64,1024])
__global__ void __launch_bounds__(128) kv_kernel(const float* __restrict__ S,
                                                 const float* __restrict__ Wv,
                                                 float* __restrict__ KV) {
  int wave = threadIdx.x >> 5;
  int b = blockIdx.x / (DD / 16);
  int e0 = (blockIdx.x % (DD / 16)) * 16;
  int m0 = wave * 16;
  const float* sb = S + (size_t)b * KK * DD + (size_t)m0 * DD;
  const float* wv = Wv + (size_t)e0 * DD;
  v8f c = {};
  for (int k0 = 0; k0 < DD; k0 += 4)
    c = wmma4(frag_contig(sb + k0, DD), frag_contig(wv + k0, DD), c);
  store_tile(KV + (size_t)b * KK * DD + (size_t)m0 * DD + e0, DD, c);
}

// K3b: T[b] = KV[b] @ W1^T   ([B,64,64])
__global__ void __launch_bounds__(128) t_kernel(const float* __restrict__ KV,
                                                const float* __restrict__ W1,
                                                float* __restrict__ T) {
  int wave = threadIdx.x >> 5;
  int b = blockIdx.x;
  int m0 = wave * 16;
  const float* kvb = KV + (size_t)b * KK * DD + (size_t)m0 * DD;
  v8f c[4] = {{}, {}, {}, {}};
  for (int k0 = 0; k0 < DD; k0 += 4) {
    v2f a = frag_contig(kvb + k0, DD);
#pragma unroll
    for (int nt = 0; nt < 4; ++nt)
      c[nt] = wmma4(a, frag_contig(W1 + (size_t)(nt * 16) * DD + k0, DD), c[nt]);
  }
#pragma unroll
  for (int nt = 0; nt < 4; ++nt)
    store_tile(T + (size_t)b * KK * KK + (size_t)m0 * KK + nt * 16, KK, c[nt]);
}

// K4: fc = A1 + Q@T (16x64 per wave, staged in LDS), out = fc @ W2^T.
// grid: B*(N/64) blocks of 128; wave w owns rows [64*rt+16w, +16).
__global__ void __launch_bounds__(128) out_kernel(
    const float* __restrict__ Q, const float* __restrict__ A1,
    const float* __restrict__ T, const float* __restrict__ W2,
    float* __restrict__ out) {
  __shared__ float fc_lds[4][16][KK + 4];
  int wave = threadIdx.x >> 5;
  int b = blockIdx.x / (NN / 64);
  int row0 = (blockIdx.x % (NN / 64)) * 64 + wave * 16;
  const float* qb = Q + (size_t)b * NN * KK + (size_t)row0 * KK;
  const float* a1b = A1 + (size_t)b * NN * KK + (size_t)row0 * KK;
  const float* tb = T + (size_t)b * KK * KK;
  // fc = A1 + Q @ T   (accumulator seeded from the A1 tile)
#pragma unroll
  for (int ct = 0; ct < 4; ++ct) {
    v8f c = load_tile(a1b + ct * 16, KK);
    for (int j0 = 0; j0 < KK; j0 += 4) {
      v2f a = frag_contig(qb + j0, KK);
      v2f bb = frag_strided(tb + (size_t)j0 * KK + ct * 16, KK);
      c = wmma4(a, bb, c);
    }
    store_tile(&fc_lds[wave][0][ct * 16], KK + 4, c);
  }
  // out = fc @ W2^T  (same-wave LDS ops are in-order, so store->load is safe)
  const float* fcp = &fc_lds[wave][0][0];
  float* ob = out + (size_t)b * NN * DD + (size_t)row0 * DD;
  for (int dt = 0; dt < DD / 16; ++dt) {
    v8f c = {};
#pragma unroll
    for (int k0 = 0; k0 < KK; k0 += 4) {
      v2f a = frag_contig(fcp + k0, KK + 4);
      v2f bb = frag_contig(W2 + (size_t)(dt * 16) * KK + k0, KK);
      c = wmma4(a, bb, c);
    }
    store_tile(ob + dt * 16, DD, c);
  }
}

extern "C" void kernel_launch(void* const* d_in, const int* in_sizes, int n_in,
                              void* d_out, int out_size, void* d_ws,
                              size_t ws_size, hipStream_t stream) {
  const float* x = (const float*)d_in[0];
  const float* Wq = (const float*)d_in[1];
  const float* Wk = (const float*)d_in[2];
  const float* Wv = (const float*)d_in[3];
  const float* W1 = (const float*)d_in[4];
  const float* W2 = (const float*)d_in[5];
  float* out = (float*)d_out;

  float* ws = (float*)d_ws;             // ~17 MB of fp32 scratch
  float* Q = ws;                        // [B,N,64]
  float* Km = Q + (size_t)BB * NN * KK; // [B,N,64]
  float* A1 = Km + (size_t)BB * NN * KK;// [B,N,64]
  float* S = A1 + (size_t)BB * NN * KK; // [B,64,D]
  float* KV = S + (size_t)BB * KK * DD; // [B,64,D]
  float* T = KV + (size_t)BB * KK * DD; // [B,64,64]

  qka_kernel<<<BB * (NN / 16), 128, 0, stream>>>(x, Wq, Wk, W1, Q, Km, A1);
  s_kernel<<<BB * (DD / 16), 128, 0, stream>>>(x, Km, S);
  kv_kernel<<<BB * (DD / 16), 128, 0, stream>>>(S, Wv, KV);
  t_kernel<<<BB, 128, 0, stream>>>(KV, W1, T);
  out_kernel<<<BB * (NN / 64), 128, 0, stream>>>(Q, A1, T, W2, out);
}